// KA_attention_heads_mod_16655883173898
// MI455X (gfx1250) — compile-verified
//
#include <hip/hip_runtime.h>
#include <hip/hip_bf16.h>

// Shapes from the reference
#define B_  8
#define H_  12
#define P_  256
#define D_  64
#define NF_ 8
#define N_       (D_ * H_ * P_)     // 196608
#define ROWS_QK  (2 * B_ * H_ * P_) // 49152
#define ROWS_1B  (B_ * H_ * P_)     // 24576
#define XL_PER_B (5 * N_)           // 983040

#define LDSW_STRIDE 65              // 64 + 1 DWORD pad per row -> conflict-free B reads
#define LDSW_BYTES  (320 * LDSW_STRIDE * 4)

typedef __attribute__((ext_vector_type(2))) float v2f;
typedef __attribute__((ext_vector_type(8))) float v8f;
typedef __attribute__((ext_vector_type(4))) unsigned int u32x4;
typedef __attribute__((ext_vector_type(8))) int i32x8;
typedef __attribute__((ext_vector_type(4))) int i32x4;

__device__ __forceinline__ float silu_f(float x) {
    // x * sigmoid(x) via raw HW ops: v_exp_f32 + v_add + v_rcp_f32 + v_mul.
    // (HIP's __fdividef is plain a/b -> IEEE div ladder; use the rcp builtin.)
#if __has_builtin(__builtin_amdgcn_rcpf)
    return x * __builtin_amdgcn_rcpf(1.0f + __expf(-x));
#else
    return x / (1.0f + __expf(-x));
#endif
}

// ---------------------------------------------------------------------------
// TDM: load a 2D f32 tensor (rows x 64, row stride 64) into LDS at offset 0,
// with pad_enable inserting 1 DWORD after every 64 DWORDs (row stride 65).
// ---------------------------------------------------------------------------
#if __has_builtin(__builtin_amdgcn_tensor_load_to_lds)
#define HAVE_TDM 1
__device__ __forceinline__ void tdm_load_w_to_lds(const float* W, unsigned rows)
{
    unsigned long long ga = (unsigned long long)(const void*)W;
    u32x4 g0;
    g0[0] = 1u;                                    // count=1, user descriptor
    g0[1] = 0u;                                    // lds_addr = 0 (dyn-LDS base)
    g0[2] = (unsigned)(ga & 0xffffffffu);          // global_addr[31:0]
    g0[3] = (unsigned)((ga >> 32) & 0x01ffffffu)   // global_addr[56:32]
          | (2u << 30);                            // type = 2 ("image")

    i32x8 g1;
    g1[0] = (int)((2u << 16)                       // data_size = 4B
          | (1u << 20)                             // pad_enable
          | (5u << 22));                           // pad_interval: 64 DWORDs
                                                   // pad_amount bits[31:25] = 0 -> 1 DWORD
    g1[1] = (int)((64u & 0xffffu) << 16);          // tensor_dim0[15:0] = 64
    g1[2] = (int)((rows & 0xffffu) << 16);         // dim0 hi=0 | tensor_dim1[15:0]
    g1[3] = (int)((64u & 0xffffu) << 16);          // dim1 hi=0 | tile_dim0 = 64
    g1[4] = (int)(rows & 0xffffu);                 // tile_dim1 = rows, tile_dim2 = 0
    g1[5] = 64;                                    // tensor_dim0_stride[31:0] = 64
    g1[6] = 0;                                     // stride hi | dim1_stride lo (unused, 2D)
    g1[7] = 0;

    i32x4 gz = {0, 0, 0, 0};
#if defined(__clang_major__) && (__clang_major__ >= 23)
    i32x8 gz8 = {0, 0, 0, 0, 0, 0, 0, 0};
    __builtin_amdgcn_tensor_load_to_lds(g0, g1, gz, gz, gz8, 0);
#else
    __builtin_amdgcn_tensor_load_to_lds(g0, g1, gz, gz, 0);
#endif
#if __has_builtin(__builtin_amdgcn_s_wait_tensorcnt)
    __builtin_amdgcn_s_wait_tensorcnt(0);
#endif
}
#else
#define HAVE_TDM 0
#endif

// ---------------------------------------------------------------------------
// GEMM fragment layout (V_WMMA_F32_16X16X4_F32, wave32):
//   A 16x4 : lane m=lane&15 holds row m; v2f = K=kl,kl+1 with kl=(lane>>4)*2
//   B 4x16 : lane n=lane&15 holds col n; v2f = K=kl,kl+1
//   C/D    : VGPR r -> M = r + 8*(lane>>4), N = lane&15
// ---------------------------------------------------------------------------

// base = silu([q;k]) @ Wb^T   (rows=49152, K=64, cols=64)
__global__ void kb_base(const float* __restrict__ q, const float* __restrict__ k,
                        const float* __restrict__ Wb, float* __restrict__ base)
{
    const int lane = threadIdx.x & 31;
    const int wave = threadIdx.x >> 5;
    const int row0 = blockIdx.x * 16;
    const int n0   = wave * 16;
    const int m    = lane & 15;
    const int kl   = (lane >> 4) * 2;

    int r = row0 + m;
    const float* src = (r < ROWS_1B) ? q : k;
    if (r >= ROWS_1B) r -= ROWS_1B;
    const float* arow = src + (size_t)r * D_;
    const float* brow = Wb + (size_t)(n0 + m) * D_;

    v8f acc = {};
    #pragma unroll
    for (int kk = 0; kk < D_; kk += 4) {
        v2f a, b;
        a.x = silu_f(arow[kk + kl]);
        a.y = silu_f(arow[kk + kl + 1]);
        b.x = brow[kk + kl];
        b.y = brow[kk + kl + 1];
        acc = __builtin_amdgcn_wmma_f32_16x16x4_f32(false, a, false, b,
                                                    (short)0, acc, false, false);
    }
    #pragma unroll
    for (int rr = 0; rr < 8; ++rr) {
        const int mr = rr + ((lane >> 4) << 3);
        base[(size_t)(row0 + mr) * D_ + n0 + m] = acc[rr];
    }
}

// xl = x @ lin_w^T + lin_b  (rows=24576, K=64, cols=320)
// W (320x64, 80KB) staged once per block into LDS via TDM (padded rows),
// A-fragment cached in registers, each wave runs 5 col tiles = 80 WMMAs.
__global__ void kb_lin(const float* __restrict__ x, const float* __restrict__ W,
                       const float* __restrict__ bias, float* __restrict__ xl)
{
    extern __shared__ float ldsW[];   // 320 * 65 floats

#if HAVE_TDM
    if (threadIdx.x < 32) {           // wave 0 issues one TDM op for the block
        tdm_load_w_to_lds(W, 320);
    }
#else
    for (int i = threadIdx.x; i < 320 * 64; i += blockDim.x) {
        ldsW[(i >> 6) * LDSW_STRIDE + (i & 63)] = W[i];
    }
#endif
    __syncthreads();

    const int lane = threadIdx.x & 31;
    const int wave = threadIdx.x >> 5;
    const int row0 = blockIdx.x * 16;
    const int m    = lane & 15;
    const int kl   = (lane >> 4) * 2;

    const float* arow = x + (size_t)(row0 + m) * D_;
    v2f afrag[16];
    #pragma unroll
    for (int s = 0; s < 16; ++s) {
        afrag[s].x = arow[s * 4 + kl];
        afrag[s].y = arow[s * 4 + kl + 1];
    }

    #pragma unroll
    for (int t = 0; t < 5; ++t) {
        const int n0 = t * 64 + wave * 16;
        const float* wrow = ldsW + (n0 + m) * LDSW_STRIDE;
        v8f acc = {};
        #pragma unroll
        for (int s = 0; s < 16; ++s) {
            v2f b;
            b.x = wrow[s * 4 + kl];
            b.y = wrow[s * 4 + kl + 1];
            acc = __builtin_amdgcn_wmma_f32_16x16x4_f32(false, afrag[s], false, b,
                                                        (short)0, acc, false, false);
        }
        const float bv = bias[n0 + m];
        #pragma unroll
        for (int rr = 0; rr < 8; ++rr) {
            const int mr = rr + ((lane >> 4) << 3);
            xl[(size_t)(row0 + mr) * (5 * D_) + n0 + m] = acc[rr] + bv;
        }
    }
}

// out = gsum @ W_out^T + 5*b_out   (rows=24576, K=64, cols=64)
__global__ void kb_out(const float* __restrict__ g, const float* __restrict__ W,
                       const float* __restrict__ bias, float* __restrict__ out)
{
    const int lane = threadIdx.x & 31;
    const int wave = threadIdx.x >> 5;
    const int row0 = blockIdx.x * 16;
    const int n0   = wave * 16;
    const int m    = lane & 15;
    const int kl   = (lane >> 4) * 2;

    const float* arow = g + (size_t)(row0 + m) * D_;
    const float* brow = W + (size_t)(n0 + m) * D_;

    v8f acc = {};
    #pragma unroll
    for (int kk = 0; kk < D_; kk += 4) {
        v2f a, b;
        a.x = arow[kk + kl];
        a.y = arow[kk + kl + 1];
        b.x = brow[kk + kl];
        b.y = brow[kk + kl + 1];
        acc = __builtin_amdgcn_wmma_f32_16x16x4_f32(false, a, false, b,
                                                    (short)0, acc, false, false);
    }
    const float bv = 5.0f * bias[n0 + m];
    #pragma unroll
    for (int rr = 0; rr < 8; ++rr) {
        const int mr = rr + ((lane >> 4) << 3);
        out[(size_t)(row0 + mr) * D_ + n0 + m] = acc[rr] + bv;
    }
}

// Sin-expansion + NF/c2 reduction + shift/base/scale fusion.
//   gsum[b,h2,pd] = (shift_sum + sum_{c2,nf2} sin(grid[c,nf]*xl[b,c,n])*coef[n,nf,c]) * ssp[d]
//                 + 5 * base[b+off, h2, pd] * sbs[d]
__global__ void kb_sin(const float* __restrict__ xl, const float* __restrict__ coef,
                       const float* __restrict__ grid, const float* __restrict__ shift,
                       const float* __restrict__ base, const float* __restrict__ ssp,
                       const float* __restrict__ sbs, float* __restrict__ gsum,
                       int baseBatchOff)
{
    const int idx = blockIdx.x * blockDim.x + threadIdx.x; // 0 .. N_-1
    const int h2 = idx >> 14;        // / 16384
    const int pd = idx & 16383;
    const int d  = pd & 63;

    float acc[B_];
    #pragma unroll
    for (int b = 0; b < B_; ++b) acc[b] = 0.0f;

    float shsum = 0.0f;
    #pragma unroll
    for (int c2 = 0; c2 < 5; ++c2)
        shsum += shift[h2 * 81920 + c2 * 16384 + pd];

    const int u0 = h2 * 40;
    for (int du = 0; du < 40; ++du) {
        const int u  = u0 + du;
        const int c  = u / 96;         // 0..4
        const int nf = (u / 12) & 7;   // 0..7
        const int n  = (u % 12) * 16384 + pd;

        if (du + 4 < 40) {             // strided coef: prefetch ahead
            const int u2 = u0 + du + 4;
            const int n2 = (u2 % 12) * 16384 + pd;
            __builtin_prefetch(&coef[(size_t)n2 * 40 + ((u2 / 12) & 7) * 5 + u2 / 96], 0, 1);
        }

        const float cf = coef[(size_t)n * 40 + nf * 5 + c];
        const float gv = grid[c * 8 + nf];
        const float* xb = xl + (size_t)c * N_ + n;
        #pragma unroll
        for (int b = 0; b < B_; ++b) {
            const float xv = xb[(size_t)b * XL_PER_B];
            acc[b] += __sinf(gv * xv) * cf;   // v_sin_f32 (TRANS32) co-executes with FMA
        }
    }

    const float sspv = ssp[d];
    const float sbv  = sbs[d];
    #pragma unroll
    for (int b = 0; b < B_; ++b) {
        const float bse = base[(size_t)(b + baseBatchOff) * N_ + idx];
        gsum[(size_t)b * N_ + idx] = (acc[b] + shsum) * sspv + 5.0f * bse * sbv;
    }
}

extern "C" void kernel_launch(void* const* d_in, const int* in_sizes, int n_in,
                              void* d_out, int out_size, void* d_ws, size_t ws_size,
                              hipStream_t stream)
{
    const float* q          = (const float*)d_in[0];
    const float* k          = (const float*)d_in[1];
    // d_in[2] = scale (unused by the reference body)
    const float* grid       = (const float*)d_in[3];
    const float* Wb         = (const float*)d_in[4];
    const float* coef_q     = (const float*)d_in[5];
    const float* coef_k     = (const float*)d_in[6];
    const float* shift_q    = (const float*)d_in[7];
    const float* shift_k    = (const float*)d_in[8];
    const float* scale_base = (const float*)d_in[9];
    const float* scale_sp   = (const float*)d_in[10];
    const float* lin_q_w    = (const float*)d_in[11];
    const float* lin_q_b    = (const float*)d_in[12];
    const float* lin_k_w    = (const float*)d_in[13];
    const float* lin_k_b    = (const float*)d_in[14];
    const float* lo_q_w     = (const float*)d_in[15];
    const float* lo_q_b     = (const float*)d_in[16];
    const float* lo_k_w     = (const float*)d_in[17];
    const float* lo_k_b     = (const float*)d_in[18];

    // Workspace layout (floats): base[16N] | xl[40N] | gsum_q[8N] | gsum_k[8N]  ~= 56.6 MB
    float* base = (float*)d_ws;
    float* xl   = base + (size_t)16 * N_;
    float* gsq  = xl   + (size_t)40 * N_;
    float* gsk  = gsq  + (size_t)8  * N_;

    const dim3 blk(128); // 4 waves

    // 1) base = silu([q;k]) @ Wb^T
    kb_base<<<dim3(ROWS_QK / 16), blk, 0, stream>>>(q, k, Wb, base);

    // 2-3) q branch (xl buffer reused by k branch afterwards; stream-ordered)
    kb_lin<<<dim3(ROWS_1B / 16), blk, LDSW_BYTES, stream>>>(q, lin_q_w, lin_q_b, xl);
    kb_sin<<<dim3(N_ / 256), dim3(256), 0, stream>>>(xl, coef_q, grid, shift_q,
                                                     base, scale_sp, scale_base, gsq, 0);

    // 4-5) k branch
    kb_lin<<<dim3(ROWS_1B / 16), blk, LDSW_BYTES, stream>>>(k, lin_k_w, lin_k_b, xl);
    kb_sin<<<dim3(N_ / 256), dim3(256), 0, stream>>>(xl, coef_k, grid, shift_k,
                                                     base, scale_sp, scale_base, gsk, 8);

    // 6-7) output projections straight into d_out (fq then fk)
    float* outp = (float*)d_out;
    kb_out<<<dim3(ROWS_1B / 16), blk, 0, stream>>>(gsq, lo_q_w, lo_q_b, outp);
    kb_out<<<dim3(ROWS_1B / 16), blk, 0, stream>>>(gsk, lo_k_w, lo_k_b, outp + (size_t)8 * N_);
}